// NaiveCollider_18459769438923
// MI455X (gfx1250) — compile-verified
//
#include <hip/hip_runtime.h>
#include <hip/hip_bf16.h>

// -------- problem constants (match reference) --------
#define NB    8192
#define LIMB  (4 * NB)     // broad-phase candidate limit (32768)
#define LIMN  NB           // narrow-phase limit (8192)
#define EPSF  1e-12f

typedef float v2f __attribute__((ext_vector_type(2)));
typedef float v8f __attribute__((ext_vector_type(8)));

// ---------------- workspace layout (int32 units) ----------------
// [0      , 32768) : cands   (flat pair index i*8192+j, or -1)
// [32768  , 33280) : bcnt    (512 broad per-block counts)
// [33280  , 33792) : boff    (512 broad exclusive offsets)
// [33792  , 33920) : ncnt    (128 narrow per-block counts)
// [33920  , 34048) : noff    (128 narrow exclusive offsets)
// total: 136192 bytes

// Compute, for one 16x16 pair tile (rows R..R+15 x cols C..C+15),
//   max( dx - rsum, -dx - rsum, dy - rsum, -dy - rsum )
// via four rank-4 f32 WMMAs. Element (i,j) overlaps iff result <= 0.
__device__ __forceinline__ v8f tile_max4(v2f ax, v2f ay,
                                         const float* __restrict__ cen,
                                         const float* __restrict__ rad,
                                         int C, int lane) {
  const int  m  = lane & 15;
  const bool lo = lane < 16;
  const int  j  = C + m;
  const float xj = cen[2 * j], yj = cen[2 * j + 1], rj = rad[j];
  // B fragment: VGPR0 = K0 (lanes 0-15) / K2 (lanes 16-31); VGPR1 = K1 / K3
  v2f bx1, bx2, by1, by2;
  bx1.x = lo ? 1.f : -1.f;  bx1.y = lo ? -xj : -rj;   // ( 1, -xj, -1, -rj)
  bx2.x = -1.f;             bx2.y = lo ?  xj : -rj;   // (-1,  xj, -1, -rj)
  by1.x = lo ? 1.f : -1.f;  by1.y = lo ? -yj : -rj;
  by2.x = -1.f;             by2.y = lo ?  yj : -rj;
  v8f z = {0.f, 0.f, 0.f, 0.f, 0.f, 0.f, 0.f, 0.f};
  v8f d1 = __builtin_amdgcn_wmma_f32_16x16x4_f32(false, ax, false, bx1, (short)0, z, false, false);
  v8f d2 = __builtin_amdgcn_wmma_f32_16x16x4_f32(false, ax, false, bx2, (short)0, z, false, false);
  v8f d3 = __builtin_amdgcn_wmma_f32_16x16x4_f32(false, ay, false, by1, (short)0, z, false, false);
  v8f d4 = __builtin_amdgcn_wmma_f32_16x16x4_f32(false, ay, false, by2, (short)0, z, false, false);
  v8f mx;
#pragma unroll
  for (int v = 0; v < 8; ++v)
    mx[v] = fmaxf(fmaxf(d1[v], d2[v]), fmaxf(d3[v], d4[v]));
  return mx;
}

// Build the per-lane A fragments (depend only on the block's row stripe).
__device__ __forceinline__ void make_a(const float* __restrict__ cen,
                                       const float* __restrict__ rad,
                                       int R, int lane, v2f& ax, v2f& ay) {
  const int  m  = lane & 15;
  const bool lo = lane < 16;
  const int  i  = R + m;
  const float xi = cen[2 * i], yi = cen[2 * i + 1], ri = rad[i];
  ax.x = lo ? xi : ri;  ax.y = 1.f;   // A row i = (xi, 1, ri, 1)
  ay.x = lo ? yi : ri;  ay.y = 1.f;   // A row i = (yi, 1, ri, 1)
}

// ---------------- init: cands = -1, out = centers ----------------
__global__ void ck_init(const float* __restrict__ cen, float* __restrict__ out,
                        int* __restrict__ cands) {
  int g = blockIdx.x * 256 + threadIdx.x;
  if (g < LIMB) {
    cands[g] = -1;
  } else {
    int k = g - LIMB;
    if (k < 2 * NB) out[k] = cen[k];
  }
}

// ---------------- broad phase: per-block hit counts ----------------
__global__ void ck_broad_count(const float* __restrict__ cen,
                               const float* __restrict__ rad,
                               int* __restrict__ bcnt) {
  __shared__ int sred[256];
  const int R = blockIdx.x * 16;
  const int t = threadIdx.x, lane = t & 31, wave = t >> 5;
  v2f ax, ay; make_a(cen, rad, R, lane, ax, ay);
  const int j = 0 /*placeholder*/;
  (void)j;
  int acc = 0;
  for (int ct = wave; ct < NB / 16; ct += 8) {
    const int C = ct * 16;
    if (C + 15 <= R) continue;                // whole tile has j <= i
    v8f mx = tile_max4(ax, ay, cen, rad, C, lane);
    const int jc = C + (lane & 15);
    const int ib = R + ((lane >> 4) << 3);
#pragma unroll
    for (int v = 0; v < 8; ++v)
      acc += ((mx[v] <= 0.f) && (jc > ib + v)) ? 1 : 0;
  }
  sred[t] = acc; __syncthreads();
  for (int off = 128; off > 0; off >>= 1) {
    if (t < off) sred[t] += sred[t + off];
    __syncthreads();
  }
  if (t == 0) bcnt[blockIdx.x] = sred[0];
}

// ---------------- serial exclusive scan (tiny: 512 / 128 entries) -------
__global__ void ck_exscan(const int* __restrict__ src, int* __restrict__ dst, int n) {
  if (threadIdx.x == 0 && blockIdx.x == 0) {
    int run = 0;
    for (int k = 0; k < n; ++k) { dst[k] = run; run += src[k]; }
  }
}

// ---------------- broad phase: ordered candidate emission ----------------
__global__ void ck_broad_emit(const float* __restrict__ cen,
                              const float* __restrict__ rad,
                              const int* __restrict__ boff,
                              int* __restrict__ cands) {
  __shared__ unsigned short bm[16][512];   // 16 rows x 512 col-tiles (16 KB)
  __shared__ int ss[256];
  const int R = blockIdx.x * 16;
  const int t = threadIdx.x, lane = t & 31, wave = t >> 5;

  unsigned int* bm32 = (unsigned int*)&bm[0][0];
  for (int k = t; k < 16 * 512 / 2; k += 256) bm32[k] = 0u;
  __syncthreads();

  v2f ax, ay; make_a(cen, rad, R, lane, ax, ay);
  for (int ct = wave; ct < NB / 16; ct += 8) {
    const int C = ct * 16;
    if (C + 15 <= R) continue;
    v8f mx = tile_max4(ax, ay, cen, rad, C, lane);
    const int jc = C + (lane & 15);
    const int ib = R + ((lane >> 4) << 3);
#pragma unroll
    for (int v = 0; v < 8; ++v) {
      const bool keep = (mx[v] <= 0.f) && (jc > ib + v);
      unsigned bal = __builtin_amdgcn_ballot_w32(keep);
      if (lane == 0) {                       // bits[15:0]=row v, [31:16]=row v+8
        bm[v][ct]     = (unsigned short)(bal & 0xFFFFu);
        bm[v + 8][ct] = (unsigned short)(bal >> 16);
      }
    }
  }
  __syncthreads();

  int running = boff[blockIdx.x];
  for (int r = 0; r < 16; ++r) {
    const int i = R + r;
    unsigned bits = (unsigned)bm[r][2 * t] | ((unsigned)bm[r][2 * t + 1] << 16);
    const int cnt = __popc(bits);
    ss[t] = cnt; __syncthreads();
    for (int off = 1; off < 256; off <<= 1) {           // inclusive Hillis-Steele
      int addv = (t >= off) ? ss[t - off] : 0;
      __syncthreads();
      ss[t] += addv;
      __syncthreads();
    }
    const int excl = ss[t] - cnt;
    const int total = ss[255];
    int idx = running + excl;
    while (bits) {
      const int p = __ffs(bits) - 1; bits &= bits - 1;
      if (idx < LIMB) cands[idx] = (i << 13) | (32 * t + p);
      ++idx;
    }
    running += total;
    __syncthreads();
  }
}

// ---------------- narrow phase predicate (identical in both passes) ------
__device__ __forceinline__ bool narrow_hit(const float* __restrict__ cen,
                                           const float* __restrict__ rad,
                                           int f, int& i, int& j,
                                           float& px, float& py) {
  const bool valid = f >= 0;
  const int sf = valid ? f : 0;
  i = sf >> 13; j = sf & 8191;
  const float dx = cen[2 * j] - cen[2 * i];
  const float dy = cen[2 * j + 1] - cen[2 * i + 1];
  const float dist = sqrtf(dx * dx + dy * dy + EPSF);
  const float depth = rad[i] + rad[j] - dist;
  px = dx / dist * depth;
  py = dy / dist * depth;
  return valid && (depth > 0.f);
}

__global__ void ck_narrow_count(const float* __restrict__ cen,
                                const float* __restrict__ rad,
                                const int* __restrict__ cands,
                                int* __restrict__ ncnt) {
  __shared__ int sred[256];
  const int t = threadIdx.x, g = blockIdx.x * 256 + t;
  int i, j; float px, py;
  sred[t] = narrow_hit(cen, rad, cands[g], i, j, px, py) ? 1 : 0;
  __syncthreads();
  for (int off = 128; off > 0; off >>= 1) {
    if (t < off) sred[t] += sred[t + off];
    __syncthreads();
  }
  if (t == 0) ncnt[blockIdx.x] = sred[0];
}

__global__ void ck_narrow_apply(const float* __restrict__ cen,
                                const float* __restrict__ rad,
                                const int* __restrict__ cands,
                                const int* __restrict__ noff,
                                float* __restrict__ out) {
  __shared__ int ss[256];
  const int t = threadIdx.x, g = blockIdx.x * 256 + t;
  int i, j; float px, py;
  const bool hit = narrow_hit(cen, rad, cands[g], i, j, px, py);
  const int cnt = hit ? 1 : 0;
  ss[t] = cnt; __syncthreads();
  for (int off = 1; off < 256; off <<= 1) {
    int addv = (t >= off) ? ss[t - off] : 0;
    __syncthreads();
    ss[t] += addv;
    __syncthreads();
  }
  const int slot = noff[blockIdx.x] + ss[t] - cnt;
  if (hit && slot < LIMN) {
    atomicAdd(&out[2 * i],     -0.5f * px);
    atomicAdd(&out[2 * i + 1], -0.5f * py);
    atomicAdd(&out[2 * j],      0.5f * px);
    atomicAdd(&out[2 * j + 1],  0.5f * py);
  }
}

extern "C" void kernel_launch(void* const* d_in, const int* in_sizes, int n_in,
                              void* d_out, int out_size, void* d_ws, size_t ws_size,
                              hipStream_t stream) {
  const float* cen = (const float*)d_in[0];   // [8192,2] f32
  const float* rad = (const float*)d_in[1];   // [8192]   f32
  float* out = (float*)d_out;                 // [8192,2] f32
  int* W     = (int*)d_ws;
  int* cands = W;
  int* bcnt  = W + 32768;
  int* boff  = W + 33280;
  int* ncnt  = W + 33792;
  int* noff  = W + 33920;

  ck_init        <<<(LIMB + 2 * NB) / 256, 256, 0, stream>>>(cen, out, cands);
  ck_broad_count <<<NB / 16, 256, 0, stream>>>(cen, rad, bcnt);
  ck_exscan      <<<1, 1, 0, stream>>>(bcnt, boff, NB / 16);
  ck_broad_emit  <<<NB / 16, 256, 0, stream>>>(cen, rad, boff, cands);
  ck_narrow_count<<<LIMB / 256, 256, 0, stream>>>(cen, rad, cands, ncnt);
  ck_exscan      <<<1, 1, 0, stream>>>(ncnt, noff, LIMB / 256);
  ck_narrow_apply<<<LIMB / 256, 256, 0, stream>>>(cen, rad, cands, noff, out);
}